// EMRouting_90022514524274
// MI455X (gfx1250) — compile-verified
//
#include <hip/hip_runtime.h>
#include <math.h>

// EM routing (matrix capsules), MI455X / gfx1250.
// One workgroup (256 threads = 8 wave32) per spatial position.
// V tile (144x32x16 f32 = 288KB) is LDS-resident for all 3 EM iterations
// (HBM reads V exactly once: ~170MB -> ~7.3us at 23.3 TB/s).
// Weighted sums over n use chained V_WMMA_F32_16X16X4_F32.
// V staging uses GLOBAL_LOAD_ASYNC_TO_LDS_B128 (ASYNCcnt-tracked, no VGPR data).

typedef __attribute__((ext_vector_type(2))) float v2f;
typedef __attribute__((ext_vector_type(8))) float v8f;
typedef __attribute__((ext_vector_type(4))) float f4;
typedef int vsi4 __attribute__((__vector_size__(4 * sizeof(int))));

#if __has_builtin(__builtin_amdgcn_global_load_async_to_lds_b128)
#define HAVE_ASYNC_LDS 1
#else
#define HAVE_ASYNC_LDS 0
#endif

constexpr int Bb = 4, OH = 12, OW = 12, KH = 3, KW = 3, IC = 16;
constexpr int OC = 32, PD = 16;
constexpr int N = KH * KW * IC;          // 144 votes per position
constexpr int NPOS = Bb * OH * OW;       // 576 positions
constexpr int VROW = OC * PD;            // 512 floats per vote row
constexpr int VROW_P = VROW + 4;         // padded LDS stride (516, 16B aligned)
constexpr int RSTR = OC + 1;             // padded R stride (33, conflict-free)
constexpr float EPSf = 1e-7f;
constexpr float LOG2PIf = 1.8378770664093453f;

// LDS layout (float offsets)
constexpr int OFF_V  = 0;
constexpr int OFF_R  = OFF_V + N * VROW_P;   // 74304
constexpr int OFF_A  = OFF_R + N * RSTR;     // +4752
constexpr int OFF_MU = OFF_A + 160;          // mu numerator -> mu (512)
constexpr int OFF_SG = OFF_MU + OC * PD;     // E[V^2] numerator -> sigma^2 (512)
constexpr int OFF_DN = OFF_SG + OC * PD;     // denom per oc (32)
constexpr int OFF_AO = OFF_DN + OC;          // a_out (32)
constexpr int OFF_L1 = OFF_AO + OC;          // log_p1 (32)
constexpr int SMEM_FLOATS = OFF_L1 + OC;     // 80336 floats = 321344 B <= 320KB

__global__ void __launch_bounds__(256, 1)
emrouting_kernel(const float* __restrict__ Vg, const float* __restrict__ Ag,
                 const float* __restrict__ BetaU, const float* __restrict__ BetaA,
                 float* __restrict__ outMu, float* __restrict__ outA,
                 float* __restrict__ outSg)
{
    extern __shared__ float sm[];
    float* Vt = sm + OFF_V;
    float* Rl = sm + OFF_R;
    float* Al = sm + OFF_A;
    float* MU = sm + OFF_MU;
    float* SG = sm + OFF_SG;
    float* DN = sm + OFF_DN;
    float* AO = sm + OFF_AO;
    float* L1 = sm + OFF_L1;

    const int pos  = blockIdx.x;
    const int tid  = threadIdx.x;
    const int lane = tid & 31;
    const int wave = tid >> 5;

    // ---- Phase 0: stage contiguous 288KB V tile into LDS (padded rows) ----
    const f4* src = reinterpret_cast<const f4*>(Vg + (size_t)pos * (N * VROW));
    for (int i = tid; i < N * VROW / 4; i += 256) {
        int n = i >> 7;          // 128 float4 per vote row
        int r = i & 127;
        f4* d4 = reinterpret_cast<f4*>(Vt + n * VROW_P + r * 4);
#if HAVE_ASYNC_LDS
        __builtin_amdgcn_global_load_async_to_lds_b128(
            (__attribute__((address_space(1))) vsi4*)(uintptr_t)(src + i),
            (__attribute__((address_space(3))) vsi4*)(uint32_t)(uintptr_t)d4,
            0, 0);
#else
        *d4 = src[i];
#endif
    }
    if (tid < N) Al[tid] = Ag[(size_t)pos * N + tid];
    for (int i = tid; i < N * OC; i += 256)
        Rl[(i >> 5) * RSTR + (i & 31)] = 1.0f / OC;
#if HAVE_ASYNC_LDS
#if __has_builtin(__builtin_amdgcn_s_wait_asynccnt)
    __builtin_amdgcn_s_wait_asynccnt(0);
#else
    asm volatile("s_wait_asynccnt 0x0" ::: "memory");
#endif
#endif
    __syncthreads();

    for (int it = 0; it < 3; ++it) {
        const float lambda = (it == 0) ? 0.01f * (1.f - 0.95f)
                           : (it == 1) ? 0.01f * (1.f - 0.9025f)
                                       : 0.01f * (1.f - 0.857375f);

        // ---- Rw = R * a (in place; R fully rewritten by softmax later) ----
        for (int i = tid; i < N * OC; i += 256) {
            int n = i >> 5, oc = i & 31;
            Rl[n * RSTR + oc] *= Al[n];
        }
        __syncthreads();

        // ---- denom[oc] = sum_n Rw (wave 0, conflict-free stride) ----
        if (tid < OC) {
            float s = 0.f;
            for (int n = 0; n < N; ++n) s += Rl[n * RSTR + tid];
            DN[tid] = s;
        }

        // ---- WMMA: mu-num and E[V^2]-num per (oc,p); wave owns 4 oc ----
        // A[16x4]: M=p, K=4 consecutive n.  B[4x16]: col0 = Rw weights.
        const int   m     = lane & 15;            // M (pose dim) / N column
        const int   hi    = lane >> 4;            // 0 -> K={0,1}, 1 -> K={2,3}
        const float mmask = (m == 0) ? 1.f : 0.f; // col-0 selector (no EXEC games)
        for (int j = 0; j < 4; ++j) {
            const int oc = wave * 4 + j;
            const float* pA = Vt + (hi * 2) * VROW_P + oc * PD + m;
            const float* pB = Rl + (hi * 2) * RSTR + oc;
            v8f accM = {0.f, 0.f, 0.f, 0.f, 0.f, 0.f, 0.f, 0.f};
            v8f accS = {0.f, 0.f, 0.f, 0.f, 0.f, 0.f, 0.f, 0.f};
            for (int c = 0; c < N / 4; ++c) {
                float a0  = pA[0];
                float a1  = pA[VROW_P];
                float rw0 = pB[0];          // all lanes load: LDS broadcast
                float rw1 = pB[RSTR];
                v2f Av  = {a0, a1};
                v2f Av2 = {a0 * a0, a1 * a1};
                v2f Bv  = {rw0 * mmask, rw1 * mmask};
                accM = __builtin_amdgcn_wmma_f32_16x16x4_f32(
                           false, Av,  false, Bv, (short)0, accM, false, false);
                accS = __builtin_amdgcn_wmma_f32_16x16x4_f32(
                           false, Av2, false, Bv, (short)0, accS, false, false);
                pA += 4 * VROW_P;
                pB += 4 * RSTR;
            }
            // column 0 of D lives in lane 0 (M=0..7) and lane 16 (M=8..15)
            if (m == 0) {
                #pragma unroll
                for (int r = 0; r < 8; ++r) {
                    MU[oc * PD + hi * 8 + r] = accM[r];
                    SG[oc * PD + hi * 8 + r] = accS[r];
                }
            }
        }
        __syncthreads();

        // ---- mu = num/(dn+eps); sigma^2 = (E2 - 2*mu*num + mu^2*dn)/(dn+eps)
        for (int i = tid; i < OC * PD; i += 256) {
            int oc = i >> 4;
            float dn  = DN[oc];
            float inv = 1.f / (dn + EPSf);
            float ms  = MU[i];
            float mu  = ms * inv;
            float sg  = (SG[i] - 2.f * mu * ms + mu * mu * dn) * inv;
            MU[i] = mu;
            SG[i] = sg;
        }
        __syncthreads();

        // ---- activation: x = lambda*(Beta_a - cost), L2-normalize over oc,
        //      sigmoid; also log_p1 (wave 0, one oc per lane) ----
        if (tid < OC) {
            float bu = BetaU[tid];
            float cs = 0.f, ls = 0.f;
            #pragma unroll
            for (int p = 0; p < PD; ++p) {
                float l = __logf(SG[tid * PD + p]);
                cs += bu + 0.5f * l;
                ls += LOG2PIf + l;
            }
            float x  = lambda * (BetaA[tid] - cs * DN[tid]);
            float xx = x * x;
            #pragma unroll
            for (int off = 16; off > 0; off >>= 1) xx += __shfl_xor(xx, off, 32);
            float nrm = fmaxf(__builtin_sqrtf(xx), 1e-12f);
            AO[tid] = 1.f / (1.f + __expf(-x / nrm));
            L1[tid] = -0.5f * ls + EPSf;
        }
        __syncthreads();

        // ---- R update via softmax over oc (skip on final iteration) ----
        if (it < 2) {
            if (tid < N) {
                const int n = tid;
                float mx = -3.4e38f;
                for (int oc = 0; oc < OC; ++oc) {
                    float s = 0.f;
                    #pragma unroll
                    for (int p = 0; p < PD; ++p) {
                        float v  = Vt[n * VROW_P + oc * PD + p];
                        float d  = v - MU[oc * PD + p];
                        float rs = __builtin_amdgcn_rcpf(2.f * SG[oc * PD + p] + EPSf);
                        s += d * d * rs;
                    }
                    float lg = __logf(AO[oc]) + L1[oc] - s;
                    Rl[n * RSTR + oc] = lg;
                    mx = fmaxf(mx, lg);
                }
                float sum = 0.f;
                for (int oc = 0; oc < OC; ++oc) {
                    float e = __expf(Rl[n * RSTR + oc] - mx);
                    Rl[n * RSTR + oc] = e;
                    sum += e;
                }
                float isum = 1.f / sum;
                for (int oc = 0; oc < OC; ++oc) Rl[n * RSTR + oc] *= isum;
            }
            __syncthreads();
        }
    }

    // ---- outputs: mu, a_out, sigma^2 (concatenated flat in d_out) ----
    for (int i = tid; i < OC * PD; i += 256) {
        outMu[(size_t)pos * OC * PD + i] = MU[i];
        outSg[(size_t)pos * OC * PD + i] = SG[i];
    }
    if (tid < OC) outA[(size_t)pos * OC + tid] = AO[tid];
}

extern "C" void kernel_launch(void* const* d_in, const int* in_sizes, int n_in,
                              void* d_out, int out_size, void* d_ws, size_t ws_size,
                              hipStream_t stream) {
    (void)in_sizes; (void)n_in; (void)d_ws; (void)ws_size; (void)out_size;
    const float* V  = (const float*)d_in[0];
    const float* a  = (const float*)d_in[1];
    const float* Bu = (const float*)d_in[2];
    const float* Ba = (const float*)d_in[3];

    float* out   = (float*)d_out;
    float* outMu = out;                             // 576*512
    float* outA  = outMu + (size_t)NPOS * OC * PD;  // 576*32
    float* outSg = outA  + (size_t)NPOS * OC;       // 576*512

    const size_t smem = (size_t)SMEM_FLOATS * sizeof(float);
    (void)hipFuncSetAttribute((const void*)emrouting_kernel,
                              hipFuncAttributeMaxDynamicSharedMemorySize,
                              (int)smem);
    emrouting_kernel<<<NPOS, 256, smem, stream>>>(V, a, Bu, Ba, outMu, outA, outSg);
}